// CrAKNVectorAttention2D_69904887710286
// MI455X (gfx1250) — compile-verified
//
#include <hip/hip_runtime.h>
#include <hip/hip_bf16.h>

// ============================================================================
// CrAKN VectorAttention2D — algebraically collapsed.
//
// reference:  weight = softmax(..., axis=-2)  -> sum_j weight[i,j,k] == 1
//             out    = einsum('ijk,ik->ik', weight, value)
//                    = value[i,k] * sum_j weight[i,j,k]
//                    = value[i,k]
//                    = (feat @ Wv.T + bv)[i,k]
//
// Everything upstream of `value` (q/k, pairwise diffs, 4 MLPs, LN, Mish,
// softmax) cancels exactly. The surviving work is one [2048,128]x[128,128]
// GEMM + bias: 67 MFLOP, ~2.1 MB of HBM traffic -> ~0.1 us at 23.3 TB/s.
// We keep full f32 precision via V_WMMA_F32_16X16X4_F32 (f32 in, f32 acc),
// matching the reference numerics up to accumulation order.
// ============================================================================

typedef __attribute__((ext_vector_type(2))) float v2f;
typedef __attribute__((ext_vector_type(8))) float v8f;

#define NROWS 2048
#define DDIM  128

// out[n,d] = sum_c feat[n,c] * Wv[d,c] + bv[d]
//
// One wave32 computes one 16x16 output tile with 32 chained
// v_wmma_f32_16x16x4_f32 ops over K=128. 4 waves per block.
//
// Fragment layouts (CDNA5 ISA 7.12.2, wave32):
//   A 16x4 f32 (2 VGPRs): M = lane&15, K = (lane>>4)*2 + v   -> float2 load
//   B 4x16 f32 (2 VGPRs): N = lane&15, K = (lane>>4)*2 + v   -> float2 load
//       B[c,n] = Wv[n,c]  (row-major Wv gives K-contiguous float2)
//   C/D 16x16 f32 (8 VGPRs): N = lane&15, M = v + (lane>>4)*8
__global__ __launch_bounds__(128)
void crakn_value_gemm_wmma(const float* __restrict__ feat,
                           const float* __restrict__ Wv,
                           const float* __restrict__ bv,
                           float* __restrict__ out) {
  const int lane  = threadIdx.x & 31;
  const int wave  = threadIdx.x >> 5;
  const int tile  = blockIdx.x * 4 + wave;
  const int tiles_n = DDIM / 16;                // 8 tiles across output columns
  const int m0 = (tile / tiles_n) * 16;         // output row block
  const int n0 = (tile % tiles_n) * 16;         // output col block

  const int idx16 = lane & 15;                  // M for A, N for B/C
  const int half  = lane >> 4;                  // K sub-select

  // Per-lane base pointers: each K-step loads a contiguous float2.
  const float* aptr = feat + (m0 + idx16) * DDIM + half * 2;     // feat[m, c..c+1]
  const float* bptr = Wv   + (n0 + idx16) * DDIM + half * 2;     // Wv[n, c..c+1] = B[c..c+1, n]

  v8f acc = {};

#pragma unroll
  for (int c0 = 0; c0 < DDIM; c0 += 4) {
    v2f a = *(const v2f*)(aptr + c0);
    v2f b = *(const v2f*)(bptr + c0);
    // 8 args: (neg_a, A, neg_b, B, c_mod, C, reuse_a, reuse_b)
    acc = __builtin_amdgcn_wmma_f32_16x16x4_f32(
        /*neg_a=*/false, a, /*neg_b=*/false, b,
        /*c_mod=*/(short)0, acc, /*reuse_a=*/false, /*reuse_b=*/false);
  }

  // Bias + store. Lane holds column n0+idx16 for rows v + half*8.
  const float bias = bv[n0 + idx16];
#pragma unroll
  for (int v = 0; v < 8; ++v) {
    const int row = v + half * 8;
    out[(m0 + row) * DDIM + (n0 + idx16)] = acc[v] + bias;
  }
}

extern "C" void kernel_launch(void* const* d_in, const int* in_sizes, int n_in,
                              void* d_out, int out_size, void* d_ws, size_t ws_size,
                              hipStream_t stream) {
  // setup_inputs() dict order:
  //  0 feat, 1 bias, 2 mul,
  //  3 Wq, 4 bq, 5 gq, 6 beq,
  //  7 Wk, 8 bk, 9 gk, 10 bek,
  // 11 Wv, 12 bv,
  // 13 Wm1, 14 bm1, 15 gm, 16 bem, 17 Wm2, 18 bm2,
  // 19 Wp1, 20 bp1, 21 gp, 22 bep, 23 Wp2, 24 bp2,
  // 25 Ww1, 26 bw1, 27 gw, 28 bew, 29 Ww2, 30 bw2
  const float* feat = (const float*)d_in[0];
  const float* Wv   = (const float*)d_in[11];
  const float* bv   = (const float*)d_in[12];
  float* out = (float*)d_out;

  const int tiles  = (NROWS / 16) * (DDIM / 16);  // 1024 tiles
  const int blocks = tiles / 4;                   // 4 waves (tiles) per block
  crakn_value_gemm_wmma<<<blocks, 128, 0, stream>>>(feat, Wv, bv, out);
}